// dfm_47725676593295
// MI455X (gfx1250) — compile-verified
//
#include <hip/hip_runtime.h>

// ---------------------------------------------------------------------------
// DeepFM one-hot embedding + FM layer for MI455X (gfx1250, wave32).
//
// Phase 1 (bandwidth-bound, 773 MB stream, ~33 us roofline @ 23.3 TB/s):
//   recover (index, value) of each one-hot row with accumulating
//   V_WMMA_F32_16X16X4_F32 triple reductions (Sum x, Sum g*x, Sum t*x),
//   exact in fp32 since each row has exactly one nonzero.
//   Unit: one global_load_b128/lane (512 B per wave = 128 elements) feeding
//   two WMMAs. Steady state is a guard-free ping-pong over whole 4-block
//   chunks (4 b128 loads in flight = 2 KB/wave); all masking lives in a cold
//   cleanup loop of <= CH+1 zero-filled blocks (zero B => WMMA adds exact 0).
// Phase 2 (tiny): gather w[idx] rows (48 MB of tables < 192 MB L2),
//   FM sum-square trick in the reference's fp32 accumulation order.
// ---------------------------------------------------------------------------

typedef __attribute__((ext_vector_type(2))) float v2f;
typedef __attribute__((ext_vector_type(8))) float v8f;

#define NFEAT 10
#define BATCH 1024
#define EMB 64
#define WAVES_P1 4
#define TPB_P1 (WAVES_P1 * 32)
#define CH 4   // 128-element blocks per chunk (4x b128 in flight)

struct Tabs {
    const float* x[NFEAT];
    const float* w[NFEAT];
    int V[NFEAT];
};

// Two accumulating WMMAs covering one 128-element block loaded as float4/lane.
// A row 0 = 1 (value), row 1 = element-offset weights g, row 2 = block id t.
__device__ __forceinline__ void wmma_pair(v8f& c, const float4 f, const float tf,
                                          const bool isRow2,
                                          const float b00, const float b01,
                                          const float b10, const float b11) {
    v2f a0, a1, bA, bB;
    a0[0] = isRow2 ? tf : b00;  a0[1] = isRow2 ? tf : b01;
    a1[0] = isRow2 ? tf : b10;  a1[1] = isRow2 ? tf : b11;
    bA[0] = f.x; bA[1] = f.y;   // B cols: elems {8n+4h, 8n+4h+1}
    bB[0] = f.z; bB[1] = f.w;   // B cols: elems {8n+4h+2, 8n+4h+3}
    c = __builtin_amdgcn_wmma_f32_16x16x4_f32(false, a0, false, bA, (short)0, c, false, false);
    c = __builtin_amdgcn_wmma_f32_16x16x4_f32(false, a1, false, bB, (short)0, c, false, false);
}

// Unguarded chunk load: 4 global_load_b128 from one per-lane base pointer,
// consecutive blocks are +512 B immediate offsets.
__device__ __forceinline__ void load_chunk(float4 (&dst)[CH], const float4* pl, int vbase) {
#pragma unroll
    for (int j = 0; j < CH; ++j) dst[j] = pl[(vbase + j) * 32];
}

__device__ __forceinline__ void proc_chunk(v8f& c, const float4 (&f)[CH], float tfc,
                                           const bool isRow2,
                                           const float b00, const float b01,
                                           const float b10, const float b11) {
#pragma unroll
    for (int j = 0; j < CH; ++j)
        wmma_pair(c, f[j], tfc + (float)j, isRow2, b00, b01, b10, b11);
}

// One block per (batch row b, feature i); 4 waves split the row's 128-elem
// blocks; LDS reduce picks the single wave/column that saw the nonzero.
__global__ __launch_bounds__(TPB_P1) void scan_onehot_wmma(Tabs t, float* ws) {
    const int b = blockIdx.x;
    const int i = blockIdx.y;
    const int V = t.V[i];                       // all V are multiples of 4
    const float* xrow = t.x[i] + (size_t)b * (size_t)V;

    const int tid  = threadIdx.x;
    const int wave = tid >> 5;
    const int lane = tid & 31;
    const int m    = lane & 15;   // A row / B-D column index n
    const int h    = lane >> 4;   // K half-select per ISA operand layout

    // Constant A rows (per lane, per WMMA variant of the pair):
    //   row 0: 1; row 1: g in {4h,4h+1} (wmma0) / {4h+2,4h+3} (wmma1); else 0.
    float b00 = 0.f, b01 = 0.f, b10 = 0.f, b11 = 0.f;
    if (m == 0)      { b00 = 1.f; b01 = 1.f; b10 = 1.f; b11 = 1.f; }
    else if (m == 1) {
        b00 = (float)(4 * h);     b01 = (float)(4 * h + 1);
        b10 = (float)(4 * h + 2); b11 = (float)(4 * h + 3);
    }
    const bool isRow2 = (m == 2);

    const int laneOff = 8 * m + 4 * h;          // float4-aligned (16B) offset
    const float4* pl = reinterpret_cast<const float4*>(xrow + laneOff);

    const int nblk = (V + 127) >> 7;            // 128-element blocks
    const int NF   = V >> 7;                    // full blocks (no masking)
    const int per  = (nblk + WAVES_P1 - 1) / WAVES_P1;
    const int u0   = wave * per;
    const int u1   = (u0 + per < nblk) ? (u0 + per) : nblk;
    const int uF   = (u1 < NF) ? u1 : NF;       // end of this wave's full blocks
    const int nfb  = (uF > u0) ? (uF - u0) : 0; // full blocks owned
    const int ncf  = nfb / CH;                  // whole (guard-free) chunks

    v8f c = {0.f, 0.f, 0.f, 0.f, 0.f, 0.f, 0.f, 0.f};

    // Guard-free ping-pong over whole chunks.
    if (ncf > 0) {
        float4 fa[CH], fb[CH];
        int k = 0;
        float tfc = (float)u0;
        load_chunk(fa, pl, u0);
        while (k + 2 < ncf) {
            load_chunk(fb, pl, u0 + (k + 1) * CH);
            proc_chunk(c, fa, tfc, isRow2, b00, b01, b10, b11);
            tfc += (float)CH; ++k;
            load_chunk(fa, pl, u0 + (k + 1) * CH);
            proc_chunk(c, fb, tfc, isRow2, b00, b01, b10, b11);
            tfc += (float)CH; ++k;
        }
        if (k + 1 < ncf) {          // two chunks left: fa loaded, one more to go
            load_chunk(fb, pl, u0 + (k + 1) * CH);
            proc_chunk(c, fa, tfc, isRow2, b00, b01, b10, b11);
            tfc += (float)CH;
            proc_chunk(c, fb, tfc, isRow2, b00, b01, b10, b11);
        } else {                    // one chunk left in fa
            proc_chunk(c, fa, tfc, isRow2, b00, b01, b10, b11);
        }
    }

    // Cold cleanup: remaining blocks [u0 + ncf*CH, u1), zero-filled masking;
    // absorbs both the <CH leftover full blocks and the partial tail block.
    {
        const int ub = u0 + ncf * CH;
        const float4 z4 = {0.f, 0.f, 0.f, 0.f};
#pragma unroll
        for (int j = 0; j < CH + 1; ++j) {
            const int v = ub + j;
            const int e = (v << 7) + laneOff;   // e % 4 == 0, V % 4 == 0
            float4 fp = z4;
            if ((v < u1) && (e < V))
                fp = *reinterpret_cast<const float4*>(xrow + e);
            wmma_pair(c, fp, (float)v, isRow2, b00, b01, b10, b11);
        }
    }

    // D layout: lanes 0-15 hold rows 0-7 in VGPR 0-7, column n = lane.
    __shared__ float sS[WAVES_P1 * 16];
    __shared__ float sI[WAVES_P1 * 16];
    if (lane < 16) {
        const float sv = c[0];   // Sum x      (value) for column n of this wave
        const float lv = c[1];   // Sum g*x    (offset within 8-group)
        const float tv = c[2];   // Sum t*x    (128-element block id)
        float idxf = 0.f;
        if (sv != 0.f) idxf = (128.f * tv + lv) / sv + 8.f * (float)m;
        sS[wave * 16 + m] = sv;
        sI[wave * 16 + m] = idxf;
    }
    __syncthreads();
    if (tid == 0) {
        float val = 0.f, idxf = 0.f;
        for (int k = 0; k < WAVES_P1 * 16; ++k) {
            if (sS[k] != 0.f) { val = sS[k]; idxf = sI[k]; }
        }
        float* o = ws + ((size_t)i * BATCH + b) * 2;
        o[0] = val;
        o[1] = idxf;
    }
}

// One block per batch row; thread d = embedding dim. Gathers the 10 embedding
// rows (hot in L2), accumulates in the reference's order, writes fm + embs.
__global__ __launch_bounds__(EMB) void fm_gather(Tabs t, const float* ws, float* out) {
    const int b = blockIdx.x;
    const int d = threadIdx.x;
    float* orow = out + (size_t)b * (size_t)((NFEAT + 1) * EMB);
    float se = 0.f, q = 0.f;
#pragma unroll
    for (int i = 0; i < NFEAT; ++i) {
        const float val  = ws[((size_t)i * BATCH + b) * 2 + 0];
        const float idxf = ws[((size_t)i * BATCH + b) * 2 + 1];
        const int   idx  = (int)(idxf + 0.5f);
        const float e    = val * t.w[i][(size_t)idx * EMB + d];
        se += e;            // matches reference accumulation order (i = 0..9)
        q  += e * e;        // val==1.0 exactly => e*e == val^2 * w^2 bit-exact
        orow[(size_t)(i + 1) * EMB + d] = e;
    }
    orow[d] = 0.5f * (se * se - q);
}

extern "C" void kernel_launch(void* const* d_in, const int* in_sizes, int n_in,
                              void* d_out, int out_size, void* d_ws, size_t ws_size,
                              hipStream_t stream) {
    (void)n_in; (void)out_size; (void)ws_size;
    Tabs t;
    for (int i = 0; i < NFEAT; ++i) {
        t.x[i] = (const float*)d_in[i];
        t.w[i] = (const float*)d_in[NFEAT + i];
        t.V[i] = in_sizes[i] / BATCH;          // x_i is (1024, V_i)
    }
    float* ws = (float*)d_ws;                  // 10*1024*2 floats = 80 KB

    scan_onehot_wmma<<<dim3(BATCH, NFEAT), TPB_P1, 0, stream>>>(t, ws);
    fm_gather<<<dim3(BATCH), EMB, 0, stream>>>(t, ws, (float*)d_out);
}